// ListFoldLoss_84112639525734
// MI455X (gfx1250) — compile-verified
//
#include <hip/hip_runtime.h>
#include <math.h>

typedef __attribute__((ext_vector_type(2))) float v2f;
typedef __attribute__((ext_vector_type(8))) float v8f;

#define NN 1024
#define HH 512
#define NT 512

__global__ void zero_kernel(float* out) { out[0] = 0.0f; }

__global__ __launch_bounds__(NT) void listfold_kernel(const float* __restrict__ scores,
                                                      const float* __restrict__ labels,
                                                      float* __restrict__ out,
                                                      int B) {
  __shared__ float skey[NN];   // labels (sort key, descending)
  __shared__ float sval[NN];   // scores carried through sort -> osc
  __shared__ float pe[NN];     // exp(osc)  -> inclusive prefix
  __shared__ float pg[NN];     // exp(-osc) -> inclusive prefix
  __shared__ float offE[64];
  __shared__ float offG[64];
  __shared__ float red[NT];

  const int b   = blockIdx.x;
  const int tid = threadIdx.x;

  // ---- load row into LDS ----
  for (int n = tid; n < NN; n += NT) {
    skey[n] = labels[b * NN + n];
    sval[n] = scores[b * NN + n];
  }
  __syncthreads();

  // ---- bitonic sort, descending by label, carrying score ----
  for (int k = 2; k <= NN; k <<= 1) {
    for (int j = k >> 1; j > 0; j >>= 1) {
      int i = ((tid & ~(j - 1)) << 1) | (tid & (j - 1));
      int p = i | j;
      float ki = skey[i], kp = skey[p];
      bool desc = ((i & k) == 0);
      bool sw = desc ? (ki < kp) : (ki > kp);
      if (sw) {
        skey[i] = kp; skey[p] = ki;
        float vi = sval[i], vp = sval[p];
        sval[i] = vp; sval[p] = vi;
      }
      __syncthreads();
    }
  }

  // ---- e = exp(osc), g = exp(-osc) ----
  for (int n = tid; n < NN; n += NT) {
    float o = sval[n];
    pe[n] = expf(o);
    pg[n] = expf(-o);
  }
  __syncthreads();

  // ---- WMMA f32 segmented inclusive scan ----
  // D(16x16) = L(16x16 lower-tri ones) x V(16x16), K accumulated as 4x (16x4)*(4x16).
  // Column c of V = segment c (16 consecutive values); row n of D = inclusive prefix.
  // Waves 0-3 scan pe (256 elems each), waves 4-7 scan pg. Wave-uniform branch => EXEC all 1s.
  {
    int wave = tid >> 5;   // wave32
    int lane = tid & 31;
    if (wave < 8) {
      float* buf = (wave < 4) ? pe : pg;
      int base = ((wave < 4) ? wave : (wave - 4)) * 256;
      int col  = lane & 15;   // segment index (N of B/D), also row M of A
      int half = lane >> 4;   // K-half selector per ISA 16x4 / 4x16 layouts
      v8f acc = {};
#pragma unroll
      for (int kk = 0; kk < 4; ++kk) {
        int r0 = 4 * kk + half * 2;          // K column/row this lane supplies
        v2f bm, am;
        bm.x = buf[base + col * 16 + r0];        // B[r0  , col]
        bm.y = buf[base + col * 16 + r0 + 1];    // B[r0+1, col]
        am.x = (col >= r0)     ? 1.0f : 0.0f;    // A[col, r0]   (lower-tri ones)
        am.y = (col >= r0 + 1) ? 1.0f : 0.0f;    // A[col, r0+1]
        acc = __builtin_amdgcn_wmma_f32_16x16x4_f32(
            false, am, false, bm, (short)0, acc, false, false);
      }
      // D layout: VGPR v, lanes 0-15 -> row v; lanes 16-31 -> row v+8
#pragma unroll
      for (int v = 0; v < 8; ++v) {
        int row = v + half * 8;
        buf[base + col * 16 + row] = acc[v];
      }
    }
  }
  __syncthreads();

  // ---- cross-segment exclusive prefix of the 64 segment totals (row 15) ----
  if (tid < 64)       offE[tid]      = pe[tid * 16 + 15];
  else if (tid < 128) offG[tid - 64] = pg[(tid - 64) * 16 + 15];
  __syncthreads();
  if (tid == 0) {
    float run = 0.0f;
    for (int s = 0; s < 64; ++s) { float t = offE[s]; offE[s] = run; run += t; }
    run = 0.0f;
    for (int s = 0; s < 64; ++s) { float t = offG[s]; offG[s] = run; run += t; }
  }
  __syncthreads();
  for (int n = tid; n < NN; n += NT) {
    pe[n] += offE[n >> 4];
    pg[n] += offG[n >> 4];
  }
  __syncthreads();

  // ---- per-j loss terms ----
  {
    int j = tid;                                   // 0..511
    float sumE  = pe[HH] - ((j > 0) ? pe[j - 1] : 0.0f);   // sum_{n=j..512} e[n]
    float T     = pg[NN - 1] - pg[HH - 2 + j];             // sum_{m>=511+j} g[m]
    float fcorr = (j == 0) ? 2.0f : (j == 1) ? 1.0f : 0.0f;
    float denom = T * sumE - fcorr;
    float term  = logf(denom) - (sval[j] - sval[NN - 1 - j]);
    red[tid] = term;
  }
  __syncthreads();
  for (int s = NT / 2; s > 0; s >>= 1) {
    if (tid < s) red[tid] += red[tid + s];
    __syncthreads();
  }
  if (tid == 0) atomicAdd(out, red[0] / (float)B);
}

extern "C" void kernel_launch(void* const* d_in, const int* in_sizes, int n_in,
                              void* d_out, int out_size, void* d_ws, size_t ws_size,
                              hipStream_t stream) {
  const float* scores = (const float*)d_in[0];   // (B, 1024, 1) f32
  const float* labels = (const float*)d_in[1];   // (B, 1024)    f32
  float* out = (float*)d_out;                    // scalar f32
  int B = in_sizes[1] / NN;
  zero_kernel<<<1, 1, 0, stream>>>(out);
  listfold_kernel<<<B, NT, 0, stream>>>(scores, labels, out, B);
}